// F0Resonance_83872121356606
// MI455X (gfx1250) — compile-verified
//
#include <hip/hip_runtime.h>
#include <math.h>

typedef float v2f __attribute__((ext_vector_type(2)));
typedef float v8f __attribute__((ext_vector_type(8)));

#define N_OCT      16
#define N_SAMPLES  32768
#define N_EVENTS   64
#define BATCH      4
#define BE_PAIRS   (BATCH * N_EVENTS)          // 256
#define TILE       4096                        // samples per block
#define TILES_PER_PAIR (N_SAMPLES / TILE)      // 8
#define BLOCK_THREADS 256                      // 8 waves (wave32)

__device__ __forceinline__ float sigmoidf_fast(float x) {
    return 1.0f / (1.0f + __expf(-x));
}

// ---------------------------------------------------------------------------
// Kernel 0: zero the per-(b,e) max-abs accumulators in workspace
// ---------------------------------------------------------------------------
__global__ void f0res_init_ws(unsigned int* wsmax) {
    int i = blockIdx.x * blockDim.x + threadIdx.x;
    if (i < BE_PAIRS) wsmax[i] = 0u;
}

// ---------------------------------------------------------------------------
// Kernel 1: oscillator bank. One block handles one (pair, sample-tile).
// Each wave produces 16 samples per WMMA chain:
//   A (16x4 f32)  = decay weights, constant over M rows
//   B (4x16 f32)  = sin((octave+1)*x) per sample column
//   D (16x16 f32) = osc[sample n] replicated over rows  -> lane reads d[0]
// K = vgpr + 2*(lane>>4) for both A and B (CDNA5 ISA 32-bit layouts).
//
// Harmonic generation: lane-half h needs S_n = sin(n*x) for
// n = 4c + 2h + {1,2}, c = 0..3 (stride 4).  Pair recurrence:
//   S_{n+4} = 2*cos(4x)*S_n - S_{n-4}
// seeded from S1..S4 -> no per-lane array, no LDS, pure registers.
// ---------------------------------------------------------------------------
__global__ void f0res_osc(const float* __restrict__ f0_in,
                          const float* __restrict__ dec_in,
                          const float* __restrict__ spc_in,
                          float* __restrict__ out,
                          unsigned int* __restrict__ wsmax) {
    const float MIN_FREQ = 20.0f / 11025.0f;
    const float MAX_FREQ = 3000.0f / 11025.0f;
    const float PI_F     = 3.14159265358979f;
    const float INV2PI   = 0.15915494309189535f;
    const float PI2_HI   = 6.28125f;                 // exact in f32
    const float PI2_LO   = 1.9353071795864769e-3f;   // 2*pi - PI2_HI

    const int pair = blockIdx.x / TILES_PER_PAIR;
    const int tile = blockIdx.x % TILES_PER_PAIR;
    const int tid  = threadIdx.x;
    const int wave = tid >> 5;
    const int lane = tid & 31;
    const int col  = lane & 15;        // N (sample within group of 16)
    const bool hi  = (lane & 16) != 0; // K-half select for A and B operands

    // ---- per-pair scalars (recomputed per thread; trivial) ----
    const float f0  = fabsf(f0_in[pair]);
    const float w0  = (MIN_FREQ + f0 * (MAX_FREQ - MIN_FREQ)) * PI_F * spc_in[pair];
    float dv        = sigmoidf_fast(sigmoidf_fast(dec_in[pair]));
    const float dec = 0.01f + dv * (0.99f * 0.95f);   // BASE + dv*resonance_factor

    // ---- A operand: weights w[o] = dec^(o+1), octave o = 4c + 2h + k ----
    const float d1 = dec;
    const float d2 = d1 * d1;
    const float d4 = d2 * d2;
    float base = hi ? (d2 * d1) : d1;                 // dec^(2h+1)
    v2f a[4];
    #pragma unroll
    for (int c = 0; c < 4; ++c) {
        a[c].x = base;          // K = 2h     -> dec^(4c+2h+1)
        a[c].y = base * d1;     // K = 2h + 1 -> dec^(4c+2h+2)
        base *= d4;
    }

    // ---- sample loop: each wave owns a contiguous 512-sample chunk ----
    const int sbase = tile * TILE + wave * (TILE / 8);
    const long long obase = (long long)pair * N_SAMPLES;
    float mx = 0.0f;

    for (int it = 0; it < (TILE / 8) / 16; ++it) {
        const int s = sbase + it * 16 + col;          // sample index (16-aligned base)
        float x = w0 * (float)(s + 1);

        // range-reduce x into ~[-pi, pi] (two-term 2*pi reduction)
        float kq = rintf(x * INV2PI);
        float r  = fmaf(-kq, PI2_HI, x);
        r        = fmaf(-kq, PI2_LO, r);

        const float s1 = __sinf(r);
        const float c1 = __cosf(r);

        // Low harmonics S_n = sin(n*x)
        const float twoc = c1 + c1;
        const float S1 = s1;
        const float S2 = twoc * s1;                   // sin 2x
        const float S3 = fmaf(twoc, S2, -S1);         // sin 3x
        const float S4 = fmaf(twoc, S3, -S2);         // sin 4x

        // cos(4x) for the stride-4 recurrence
        const float c2x = fmaf(2.0f * c1, c1, -1.0f); // cos 2x
        const float c4x = fmaf(2.0f * c2x, c2x, -1.0f);
        const float t4  = c4x + c4x;                  // 2*cos(4x)

        // Seed pair per lane-half:  h=0: n={1,2}, prev n={-3,-2}
        //                           h=1: n={3,4}, prev n={-1, 0}
        float p0 = hi ? S3 : S1;
        float p1 = hi ? S4 : S2;
        float q0 = hi ? -S1 : -S3;
        float q1 = hi ? 0.0f : -S2;

        // 4 chained WMMA f32 16x16x4 ops: D += A_c x B_c  (K total = 16)
        v8f dmat = {0.f, 0.f, 0.f, 0.f, 0.f, 0.f, 0.f, 0.f};
        #pragma unroll
        for (int c = 0; c < 4; ++c) {
            v2f b;
            b.x = p0;
            b.y = p1;
            dmat = __builtin_amdgcn_wmma_f32_16x16x4_f32(
                false, a[c], false, b, (short)0, dmat, false, false);
            // advance pair: S_{n+4} = t4*S_n - S_{n-4}
            float n0 = fmaf(t4, p0, -q0);
            float n1 = fmaf(t4, p1, -q1);
            q0 = p0; q1 = p1;
            p0 = n0; p1 = n1;
        }

        // D[m][n] = osc[sample n] for every m -> lane's d[0] is its own sample
        const float v = dmat[0];
        if (!hi) {
            out[obase + s] = v;
            mx = fmaxf(mx, fabsf(v));
        }
    }

    // ---- block reduction of max|osc|, then atomicMax into workspace ----
    __shared__ float red[BLOCK_THREADS];
    red[tid] = mx;
    __syncthreads();
    #pragma unroll
    for (int off = BLOCK_THREADS / 2; off > 0; off >>= 1) {
        if (tid < off) red[tid] = fmaxf(red[tid], red[tid + off]);
        __syncthreads();
    }
    if (tid == 0) {
        // non-negative floats: uint bit pattern is order-preserving
        atomicMax(&wsmax[pair], __float_as_uint(red[0]));
    }
}

// ---------------------------------------------------------------------------
// Kernel 2: max-norm  out[i] /= (max|osc| + 1e-8)
// ---------------------------------------------------------------------------
__global__ void f0res_norm(float* __restrict__ out,
                           const unsigned int* __restrict__ wsmax) {
    const int i = blockIdx.x * blockDim.x + threadIdx.x;   // < 2^23, fits int
    const int pair = i >> 15;                              // / N_SAMPLES
    const float mxv = __uint_as_float(wsmax[pair]);
    out[i] = out[i] / (mxv + 1e-8f);
}

// ---------------------------------------------------------------------------
extern "C" void kernel_launch(void* const* d_in, const int* in_sizes, int n_in,
                              void* d_out, int out_size, void* d_ws, size_t ws_size,
                              hipStream_t stream) {
    const float* f0   = (const float*)d_in[0];   // (4,64,1)
    const float* decc = (const float*)d_in[1];   // (4,64,1)
    // d_in[2] = phase_offsets: computed but unused in reference output
    const float* spc  = (const float*)d_in[3];   // (4,64,1)
    float* out = (float*)d_out;                  // (4,64,32768)
    unsigned int* wsmax = (unsigned int*)d_ws;   // 256 entries

    f0res_init_ws<<<1, BE_PAIRS, 0, stream>>>(wsmax);

    const int grid1 = BE_PAIRS * TILES_PER_PAIR;           // 2048 blocks
    f0res_osc<<<grid1, BLOCK_THREADS, 0, stream>>>(f0, decc, spc, out, wsmax);

    const int total = BE_PAIRS * N_SAMPLES;                // 8388608
    f0res_norm<<<total / 256, 256, 0, stream>>>(out, wsmax);
}